// SpatioTemporalGaussianProcessAttention_9972914061580
// MI455X (gfx1250) — compile-verified
//
#include <hip/hip_runtime.h>
#include <math.h>

#define C_ 64
#define T_ 8
#define B_ 4
#define HW_ 9216
#define NC_ 4

// workspace layout (float offsets)
#define OFF_K     0        // 4096  K[c][s][t]
#define OFF_A     4096     // 4096  folded projection (bnscale * proj/sigma)
#define OFF_MASK  8192     // 8
#define OFF_BVEC  8256     // 64
#define OFF_TT    8320     // 1
#define OFF_G     8448     // 256   per (b,c) global mean
#define OFF_COVW  8704     // 256   sigmoid gate
#define OFF_FSUM  8960     // 2048  spatial sums of fused per (b,c,t)
#define OFF_BIAS2 11008    // 2048  per (b,t) folded bias (A@pm + bvec)
#define OFF_FUSED 16384    // B*C*T*HW = 18,874,368 floats

typedef float v2f __attribute__((ext_vector_type(2)));
typedef float v8f __attribute__((ext_vector_type(8)));

__device__ __forceinline__ float gelu_f(float v) {
  return 0.5f * v * (1.0f + erff(v * 0.7071067811865475f));
}

// ---------------------------------------------------------------- zero
__global__ void zero_fsum_kernel(float* __restrict__ ws) {
  int j = blockIdx.x * 256 + threadIdx.x;
  if (j < 2048) ws[OFF_FSUM + j] = 0.0f;
}

// ---------------------------------------------------------------- prep
// softmax(kernel_weights), RBFs, K[c][s][t], temporal mask, tanh(temp),
// power-iteration spectral norm of proj_w, folded A and bias vector.
__global__ __launch_bounds__(64) void prep_kernel(
    const float* __restrict__ tk, const float* __restrict__ kw,
    const float* __restrict__ lsc, const float* __restrict__ tb,
    const float* __restrict__ prj, const float* __restrict__ bng,
    const float* __restrict__ bnb, const float* __restrict__ bnm,
    const float* __restrict__ bnv, const float* __restrict__ temp,
    float* __restrict__ ws) {
  int tid = threadIdx.x;
  __shared__ float swts[NC_];
  __shared__ float srbf[NC_ * 64];
  __shared__ float sv[64], su[64];
  __shared__ float smk[8];
  __shared__ float snorm;

  if (tid == 0) {
    float mx = kw[0];
    for (int i = 1; i < NC_; ++i) mx = fmaxf(mx, kw[i]);
    float s = 0.0f, e[NC_];
    for (int i = 0; i < NC_; ++i) { e[i] = expf(kw[i] - mx); s += e[i]; }
    for (int i = 0; i < NC_; ++i) swts[i] = e[i] / s;
  }
  __syncthreads();
  for (int j = tid; j < NC_ * 64; j += 64) {
    int i = j >> 6, ss = (j >> 3) & 7, t = j & 7;
    float ls = expf(lsc[i]);
    float dt = (float)(ss - t);
    srbf[j] = expf(-dt * dt / (2.0f * ls * ls));
  }
  __syncthreads();
  for (int st = 0; st < 64; ++st) {            // K row for c = tid
    int ss = st >> 3, t = st & 7;
    float k = 0.0f;
    for (int i = 0; i < NC_; ++i)
      k += swts[i] * tk[i * 64 + tid] * srbf[i * 64 + ss * 8 + t];
    ws[OFF_K + tid * 64 + st] = k;
  }
  if (tid < 8) {                               // mean over channels
    float m = 0.0f;
    for (int c = 0; c < 64; ++c) m += tb[c * 8 + tid];
    smk[tid] = m / 64.0f;
  }
  __syncthreads();
  if (tid == 0) {
    float mx = smk[0];
    for (int t = 1; t < 8; ++t) mx = fmaxf(mx, smk[t]);
    float s = 0.0f, e[8];
    for (int t = 0; t < 8; ++t) { e[t] = expf(smk[t] - mx); s += e[t]; }
    for (int t = 0; t < 8; ++t) ws[OFF_MASK + t] = e[t] / s;
    ws[OFF_TT] = tanhf(temp[0]);
  }
  // power iteration: sigma = sqrt(lambda_max(P^T P))
  sv[tid] = 0.125f;
  __syncthreads();
  float lam = 1.0f;
  for (int it = 0; it < 64; ++it) {
    float u = 0.0f;
    for (int c = 0; c < 64; ++c) u += prj[tid * 64 + c] * sv[c];
    su[tid] = u;
    __syncthreads();
    float w = 0.0f;
    for (int o = 0; o < 64; ++o) w += prj[o * 64 + tid] * su[o];
    sv[tid] = w;
    __syncthreads();
    if (tid == 0) {
      float n = 0.0f;
      for (int c = 0; c < 64; ++c) n += sv[c] * sv[c];
      snorm = sqrtf(n);
    }
    __syncthreads();
    lam = snorm;
    float rn = (lam > 0.0f) ? (1.0f / lam) : 0.0f;
    sv[tid] *= rn;
    __syncthreads();
  }
  float sig = sqrtf(lam);
  float rsig = (sig > 0.0f) ? (1.0f / sig) : 1.0f;
  float bsc = bng[tid] * rsqrtf(bnv[tid] + 1e-5f);   // fold BN into projection
  for (int c = 0; c < 64; ++c)
    ws[OFF_A + tid * 64 + c] = bsc * prj[tid * 64 + c] * rsig;
  ws[OFF_BVEC + tid] = bnb[tid] - bsc * bnm[tid];
}

// ---------------------------------------------------------------- global means
__global__ __launch_bounds__(256) void gmean_kernel(const float* __restrict__ x,
                                                    float* __restrict__ ws) {
  int bc = blockIdx.x;
  int b = bc >> 6, c = bc & 63;
  int tid = threadIdx.x;
  float s = 0.0f;
  for (int t = 0; t < T_; ++t) {
    const float* base = x + ((size_t)(b * T_ + t) * C_ + c) * HW_;
    for (int j = tid; j < HW_; j += 256) s += base[j];
  }
  __shared__ float red[256];
  red[tid] = s;
  __syncthreads();
  for (int k = 128; k > 0; k >>= 1) {
    if (tid < k) red[tid] += red[tid + k];
    __syncthreads();
  }
  if (tid == 0) ws[OFF_G + bc] = red[0] / (float)(T_ * HW_);
}

// ---------------------------------------------------------------- gate MLP
__global__ __launch_bounds__(256) void covw_kernel(
    const float* __restrict__ sw1, const float* __restrict__ sb1,
    const float* __restrict__ sw2, const float* __restrict__ sb2,
    float* __restrict__ ws) {
  int tid = threadIdx.x;
  int b = tid >> 6, c = tid & 63;
  float h1[8];
  for (int j = 0; j < 8; ++j) {
    float s = sb1[j];
    for (int c2 = 0; c2 < 64; ++c2) s += sw1[j * 64 + c2] * ws[OFF_G + b * 64 + c2];
    h1[j] = gelu_f(s);
  }
  float s = sb2[64 + c];
  for (int j = 0; j < 8; ++j) s += sw2[(64 + c) * 8 + j] * h1[j];
  ws[OFF_COVW + tid] = 1.0f / (1.0f + expf(-s));
}

// ---------------------------------------------------------------- big fused kernel
// Per block: 16 pixels x 64 channels x 8 timesteps staged in LDS.
// GP temporal mix + mask blend + depthwise conv3 + BN + GELU, then the
// per-pixel uncertainty MLP hidden layer as a WMMA GEMM (16o x 64c x 128
// (p,t)-columns = 8 waves x one 16x16 tile, tile w <-> t=w), fused output
// to ws, spatial sums accumulated for the (algebraically pooled) conv.
#define SB(c, t, p) (((c) * 8 + (t)) * 17 + (p))

__global__ __launch_bounds__(256) void fused_kernel(
    const float* __restrict__ x,
    const float* __restrict__ dwp, const float* __restrict__ dg,
    const float* __restrict__ dbta, const float* __restrict__ dmn,
    const float* __restrict__ dvr,
    const float* __restrict__ uw1, const float* __restrict__ ub1,
    const float* __restrict__ uw2, const float* __restrict__ ub2,
    float* __restrict__ ws) {
  __shared__ float sK[64 * 65];
  __shared__ float sbuf[64 * 8 * 17];
  __shared__ __align__(16) float su1[16 * 68];   // unc_w1, padded rows
  __shared__ float sfa[8 * 16], sfb[8 * 16];
  __shared__ float scov[64], sdsc[64], sdsh[64], sdw[64 * 3];
  __shared__ float smask[8];

  int tid = threadIdx.x;
  int b = blockIdx.x / 576;
  int hw0 = (blockIdx.x % 576) * 16;

  for (int j = tid; j < 4096; j += 256) sK[(j >> 6) * 65 + (j & 63)] = ws[OFF_K + j];
  for (int j = tid; j < 1024; j += 256) su1[(j >> 6) * 68 + (j & 63)] = uw1[j];
  if (tid < 64) {
    float sc = dg[tid] * rsqrtf(dvr[tid] + 1e-5f);
    sdsc[tid] = sc;
    sdsh[tid] = dbta[tid] - sc * dmn[tid];
    scov[tid] = ws[OFF_COVW + b * 64 + tid];
    sdw[tid * 3 + 0] = dwp[tid * 3 + 0];
    sdw[tid * 3 + 1] = dwp[tid * 3 + 1];
    sdw[tid * 3 + 2] = dwp[tid * 3 + 2];
  }
  if (tid < 8) smask[tid] = ws[OFF_MASK + tid];
  for (int j = tid; j < 8192; j += 256) {      // coalesced tile load
    int p = j & 15, c = (j >> 4) & 63, t = j >> 10;
    sbuf[SB(c, t, p)] = x[((size_t)(b * T_ + t) * C_ + c) * HW_ + hw0 + p];
  }
  __syncthreads();

  int c = tid >> 2, pg = tid & 3;
  {
    float w0 = sdw[c * 3 + 0], w1 = sdw[c * 3 + 1], w2 = sdw[c * 3 + 2];
    float dsc = sdsc[c], dsh = sdsh[c], cw = scov[c];
    for (int pp = 0; pp < 4; ++pp) {
      int p = pg * 4 + pp;
      float xv[8], f0[8];
#pragma unroll
      for (int t = 0; t < 8; ++t) xv[t] = sbuf[SB(c, t, p)];
#pragma unroll
      for (int s = 0; s < 8; ++s) {
        float g = 0.0f;
#pragma unroll
        for (int t = 0; t < 8; ++t) g += sK[c * 65 + s * 8 + t] * xv[t];
        f0[s] = cw * g + (1.0f - cw) * smask[s] * xv[s];
      }
#pragma unroll
      for (int t = 0; t < 8; ++t) {            // depthwise conv3 (pad 1) + BN + GELU
        float dd = w1 * f0[t];
        if (t > 0) dd += w0 * f0[t - 1];
        if (t < 7) dd += w2 * f0[t + 1];
        dd = dsc * dd + dsh;
        sbuf[SB(c, t, p)] = gelu_f(dd);
      }
    }
  }
  __syncthreads();
  {
    // Uncertainty MLP hidden layer as WMMA: D[o][col] = sum_c uw1[o][c]*d[c][col]
    // Wave w owns columns (t=w, p=0..15).  A-frag from padded su1 (bank-clean),
    // B-frag from sbuf (half-waves hit disjoint bank ranges).
    int wave = tid >> 5, lane = tid & 31;
    int lane16 = lane & 15, hi = lane >> 4;
    int kofs = hi ? 2 : 0;                     // lanes16-31 carry K{2,3} (ISA layout)
    v8f hidden = {};
#pragma unroll
    for (int k0 = 0; k0 < 64; k0 += 4) {
      v2f a = *(const v2f*)(su1 + lane16 * 68 + k0 + kofs);
      v2f bb;
      bb.x = sbuf[SB(k0 + kofs, wave, lane16)];
      bb.y = sbuf[SB(k0 + kofs + 1, wave, lane16)];
      hidden = __builtin_amdgcn_wmma_f32_16x16x4_f32(
          /*neg_a=*/false, a, /*neg_b=*/false, bb,
          /*c_mod=*/(short)0, hidden, /*reuse_a=*/false, /*reuse_b=*/false);
    }
    // gelu + second layer (2x16) reduced over M: 8 regs here, partner half-wave
    // holds o+8; one shfl_xor(16) completes the 16-term sum.
    float accmu = 0.0f, accva = 0.0f;
#pragma unroll
    for (int r = 0; r < 8; ++r) {
      int o = r + (hi ? 8 : 0);
      float u1v = gelu_f(hidden[r] + ub1[o]);
      accmu += uw2[o] * u1v;
      accva += uw2[16 + o] * u1v;
    }
    accmu += __shfl_xor(accmu, 16, 32);
    accva += __shfl_xor(accva, 16, 32);
    if (!hi) {
      accmu += ub2[0];
      accva += ub2[1];
      float prec = 1.0f / (accva + 1e-6f);
      float inv = 1.0f / (prec + 1.0f);
      sfa[wave * 16 + lane16] = accmu * prec * inv;   // fused = fa + d * fb
      sfb[wave * 16 + lane16] = inv;
    }
  }
  __syncthreads();
  float tsum[8];
#pragma unroll
  for (int t = 0; t < 8; ++t) tsum[t] = 0.0f;
  for (int pp = 0; pp < 4; ++pp) {
    int p = pg * 4 + pp;
#pragma unroll
    for (int t = 0; t < 8; ++t) {
      float f = sfa[t * 16 + p] + sbuf[SB(c, t, p)] * sfb[t * 16 + p];
      ws[OFF_FUSED + (((size_t)b * C_ + c) * T_ + t) * HW_ + hw0 + p] = f;
      tsum[t] += f;
    }
  }
#pragma unroll
  for (int t = 0; t < 8; ++t) {                // wave32 shuffle reduce over 4 lanes
    float v = tsum[t];
    v += __shfl_xor(v, 1, 32);
    v += __shfl_xor(v, 2, 32);
    if (pg == 0) atomicAdd(&ws[OFF_FSUM + ((size_t)b * C_ + c) * T_ + t], v);
  }
}

// ---------------------------------------------------------------- pooled conv + folded bias
// p.mean(h,w) == conv(fused.mean(h,w)), then fold A@pm + bvec into bias2[b][t][o].
__global__ __launch_bounds__(256) void poolbias_kernel(
    const float* __restrict__ pwp, const float* __restrict__ pbp,
    float* __restrict__ ws) {
  __shared__ float sfm[2048];
  __shared__ float spm[2048];
  int tid = threadIdx.x;
  for (int j = tid; j < 2048; j += 256) sfm[j] = ws[OFF_FSUM + j] * (1.0f / (float)HW_);
  __syncthreads();
  for (int j = tid; j < 2048; j += 256) {
    int b = j >> 9, o = (j >> 3) & 63, t = j & 7;
    float s = pbp[o];
    for (int cc = 0; cc < 64; ++cc) {
      const float* w = pwp + (o * 64 + cc) * 3;
      const float* f = sfm + (b * 64 + cc) * 8;
      if (t > 0) s += w[0] * f[t - 1];
      s += w[1] * f[t];
      if (t < 7) s += w[2] * f[t + 1];
    }
    spm[(b * 64 + o) * 8 + t] = s;
  }
  __syncthreads();
  for (int j = tid; j < 2048; j += 256) {
    int o = j & 63, t = (j >> 6) & 7, b = j >> 9;
    float s = ws[OFF_BVEC + o];
    const float* arow = ws + OFF_A + o * 64;
    for (int cc = 0; cc < 64; ++cc) s += arow[cc] * spm[(b * 64 + cc) * 8 + t];
    ws[OFF_BIAS2 + ((size_t)b * 8 + t) * 64 + o] = s;
  }
}

// ---------------------------------------------------------------- WMMA projection
// y[o,pix] = sum_c A[o][c] * fused[b,c,t,pix] + bias2[b,t,o];
// out = x + tanh(temp)*y.  Full-precision V_WMMA_F32_16X16X4_F32,
// one 16(o) x 16(px) tile per wave, K=64 in 16 WMMA steps.
__global__ __launch_bounds__(256) void proj_kernel(const float* __restrict__ x,
                                                   float* __restrict__ out,
                                                   const float* __restrict__ ws) {
  __shared__ __align__(16) float sA[64 * 68];  // padded: conflict-free A-frag reads
  __shared__ float sb2[64];
  int tid = threadIdx.x;
  int bt = blockIdx.x / 72;
  int tb = blockIdx.x % 72;
  for (int j = tid; j < 4096; j += 256) sA[(j >> 6) * 68 + (j & 63)] = ws[OFF_A + j];
  if (tid < 64) sb2[tid] = ws[OFF_BIAS2 + (size_t)bt * 64 + tid];
  __syncthreads();

  int b = bt >> 3, t = bt & 7;
  int wave = tid >> 5, lane = tid & 31;
  int lane16 = lane & 15;
  int hi = (lane >> 4) & 1;
  int kofs = hi ? 2 : 0;                       // ISA A 16x4 layout: lanes16-31 hold K{2,3}
  int px0 = tb * 128 + wave * 16;

  const float* fbase = ws + OFF_FUSED + (((size_t)b * C_) * T_ + t) * (size_t)HW_
                       + px0 + lane16;
  const size_t cstride = (size_t)T_ * HW_;

  v8f acc = {};
#pragma unroll
  for (int k0 = 0; k0 < 64; k0 += 4) {
    v2f a = *(const v2f*)(sA + lane16 * 68 + k0 + kofs);   // A[m=lane16][k0+kofs..+1]
    v2f bb;
    bb.x = fbase[(size_t)(k0 + kofs) * cstride];           // B[k][n=lane16]
    bb.y = fbase[(size_t)(k0 + kofs + 1) * cstride];
    acc = __builtin_amdgcn_wmma_f32_16x16x4_f32(
        /*neg_a=*/false, a, /*neg_b=*/false, bb,
        /*c_mod=*/(short)0, acc, /*reuse_a=*/false, /*reuse_b=*/false);
  }

  float tt = ws[OFF_TT];
#pragma unroll
  for (int r = 0; r < 8; ++r) {                // C/D layout: VGPR r -> M=r (lo) / r+8 (hi)
    int o = r + (hi ? 8 : 0);
    size_t idx = ((size_t)bt * C_ + o) * HW_ + px0 + lane16;
    float y = acc[r] + sb2[o];
    out[idx] = x[idx] + tt * y;
  }
}

// ---------------------------------------------------------------- launch
extern "C" void kernel_launch(void* const* d_in, const int* in_sizes, int n_in,
                              void* d_out, int out_size, void* d_ws, size_t ws_size,
                              hipStream_t stream) {
  (void)in_sizes; (void)n_in; (void)out_size; (void)ws_size;
  const float* x   = (const float*)d_in[0];
  const float* tk  = (const float*)d_in[2];
  const float* kw  = (const float*)d_in[3];
  const float* lsc = (const float*)d_in[4];
  const float* tb  = (const float*)d_in[5];
  const float* sw1 = (const float*)d_in[6];
  const float* sb1 = (const float*)d_in[7];
  const float* sw2 = (const float*)d_in[8];
  const float* sb2 = (const float*)d_in[9];
  const float* dw  = (const float*)d_in[10];
  const float* dg  = (const float*)d_in[11];
  const float* db  = (const float*)d_in[12];
  const float* dm  = (const float*)d_in[13];
  const float* dv  = (const float*)d_in[14];
  const float* uw1 = (const float*)d_in[15];
  const float* ub1 = (const float*)d_in[16];
  const float* uw2 = (const float*)d_in[17];
  const float* ub2 = (const float*)d_in[18];
  const float* pw  = (const float*)d_in[19];
  const float* pb  = (const float*)d_in[20];
  const float* prj = (const float*)d_in[21];
  const float* bg  = (const float*)d_in[22];
  const float* bb  = (const float*)d_in[23];
  const float* bm  = (const float*)d_in[24];
  const float* bv  = (const float*)d_in[25];
  const float* tmp = (const float*)d_in[26];
  float* ws  = (float*)d_ws;
  float* out = (float*)d_out;

  zero_fsum_kernel<<<8, 256, 0, stream>>>(ws);
  prep_kernel<<<1, 64, 0, stream>>>(tk, kw, lsc, tb, prj, bg, bb, bm, bv, tmp, ws);
  gmean_kernel<<<256, 256, 0, stream>>>(x, ws);
  covw_kernel<<<1, 256, 0, stream>>>(sw1, sb1, sw2, sb2, ws);
  fused_kernel<<<B_ * 576, 256, 0, stream>>>(x, dw, dg, db, dm, dv,
                                             uw1, ub1, uw2, ub2, ws);
  poolbias_kernel<<<1, 256, 0, stream>>>(pw, pb, ws);
  proj_kernel<<<B_ * T_ * 72, 256, 0, stream>>>(x, out, ws);
}